// AttentionLayer_2482491097341
// MI455X (gfx1250) — compile-verified
//
#include <hip/hip_runtime.h>
#include <math.h>

// MI455X (gfx1250) implementation.
// Memory-bound problem (~0.45 FLOP/byte): stream node_out once via coalesced
// float4 loads -> LDS (transposed, padded layout), do the per-head 32x32
// mat-vec contraction with chained V_WMMA_F32_16X16X4_F32 (B = attention
// vector replicated across the 16 N columns), softmax-ish phase fully in
// LDS/registers. One 256-thread block (8 wave32) per (b,l) site.

typedef float v2f __attribute__((ext_vector_type(2)));
typedef float v8f __attribute__((ext_vector_type(8)));

#define EPS_F 1e-6f
#define KN   32   // neighbors
#define NFG  4
#define NH   16   // heads
#define NFO  32
#define ROWLEN 258  // LDS row stride (floats) for no_s: 16*16 + 2 pad (8B aligned,
                    // odd/2 stride => conflict-free 16-lane gathers at stride ROWLEN)

__global__ __launch_bounds__(256)
void attn_layer_kernel(const float* __restrict__ beta,
                       const float* __restrict__ sa,
                       const float* __restrict__ ac,
                       const float* __restrict__ no,
                       const float* __restrict__ gw,
                       float* __restrict__ out,       // [sites*512] output part
                       float* __restrict__ out_attn)  // [sites*512] attn_final part
{
    __shared__ float no_s[NFO * ROWLEN];  // half-K tile of node_out, layout [f][k_local*16 + h]
    __shared__ float attn_s[KN * NH];     // attention [k][h]
    __shared__ float mx_s[16 * NH];       // partial maxima [k2][h]

    const int site = blockIdx.x;
    const int tid  = threadIdx.x;
    const int h    = tid & 15;     // head this thread owns in phase 1
    const int k2   = tid >> 4;     // 0..15: neighbor pair {k2, k2+16}

    const float* ac_p = ac   + (size_t)site * (KN * NFG * NH);
    const float* no_p = no   + (size_t)site * (KN * NFO * NH);
    const float* bt_p = beta + (size_t)site * (NFG * NH);
    const float* sa_p = sa   + (size_t)site * (NFG * NH);
    const float* gw_p = gw   + (size_t)site * (KN * NFG);

    // ---------------- Phase 1: attention weights ----------------
    // t[k,f,h] = (ac + (k==0)*sa) * (beta + eps); softmax-stabilize over (K,nfg) per h
    float t[8];
    float g[8];
    float lmax = -3.402823466e38f;
    #pragma unroll
    for (int j = 0; j < 2; ++j) {
        const int k = k2 + j * 16;
        #pragma unroll
        for (int f = 0; f < NFG; ++f) {
            float v = ac_p[k * (NFG * NH) + f * NH + h];
            if (k == 0) v += sa_p[f * NH + h];
            v *= (bt_p[f * NH + h] + EPS_F);
            t[j * 4 + f] = v;
            g[j * 4 + f] = gw_p[k * NFG + f];
            lmax = fmaxf(lmax, v);
        }
    }
    mx_s[k2 * NH + h] = lmax;
    __syncthreads();
    float m = mx_s[h];
    #pragma unroll
    for (int i = 1; i < 16; ++i) m = fmaxf(m, mx_s[i * NH + h]);

    float a0 = 0.f, a1 = 0.f;
    #pragma unroll
    for (int f = 0; f < 4; ++f) a0 += __expf(t[f] - m) * g[f];
    #pragma unroll
    for (int f = 0; f < 4; ++f) a1 += __expf(t[4 + f] - m) * g[4 + f];
    attn_s[k2 * NH + h]        = a0;
    attn_s[(k2 + 16) * NH + h] = a1;
    __syncthreads();
    float denom = EPS_F;
    #pragma unroll
    for (int k = 0; k < KN; ++k) denom += fabsf(attn_s[k * NH + h]);
    __syncthreads();                 // everyone done reading before overwrite
    const float rden = 1.0f / denom;
    a0 *= rden; a1 *= rden;
    attn_s[k2 * NH + h]        = a0;
    attn_s[(k2 + 16) * NH + h] = a1;
    out_attn[(size_t)site * 512 + k2 * NH + h]        = a0;
    out_attn[(size_t)site * 512 + (k2 + 16) * NH + h] = a1;
    __syncthreads();

    // ---------------- Phase 2: WMMA contraction over K ----------------
    const int lane = tid & 31;
    const int wave = tid >> 5;     // 0..7; each wave owns 4 (head, f-tile) tasks
    const int mrow = lane & 15;    // A-matrix row within lane half
    const int sel  = lane >> 4;    // 0/1: which K sub-pair this lane half holds

    v8f acc[4] = {};

    #pragma unroll
    for (int half = 0; half < 2; ++half) {
        // Stage 16 neighbors (8192 floats = 32KB) of node_out into LDS,
        // transposed to [f][k_local*16 + h] with padded rows.
        #pragma unroll
        for (int j = 0; j < 8; ++j) {
            const int idx4 = (j * 256 + tid) * 4;                // 0..8191, 16B aligned
            const float4 v = *(const float4*)(no_p + half * 8192 + idx4);
            const int kl = idx4 >> 9;          // local k (0..15)
            const int f  = (idx4 >> 4) & 31;   // nfo row
            const int hh = idx4 & 15;          // head (multiple of 4)
            float* dst = &no_s[f * ROWLEN + kl * 16 + hh];
            ((float2*)dst)[0] = make_float2(v.x, v.y);
            ((float2*)dst)[1] = make_float2(v.z, v.w);
        }
        __syncthreads();

        #pragma unroll
        for (int i = 0; i < 4; ++i) {
            const int task  = wave * 4 + i;
            const int th    = task & 15;       // head (uniform per wave)
            const int ftile = task >> 4;       // 0/1: which 16 of the 32 nfo rows
            const int rowbase = (ftile * 16 + mrow) * ROWLEN;
            #pragma unroll
            for (int kc = 0; kc < 4; ++kc) {
                const int klb = kc * 4 + sel * 2;        // local K for this lane half
                v2f a, b;
                a.x = no_s[rowbase + klb * 16 + th];     // A[m, 2*sel]   = no[k, f, h]
                a.y = no_s[rowbase + (klb + 1) * 16 + th];
                const int kg = half * 16 + klb;
                b.x = attn_s[kg * NH + th];              // B[2*sel,  n] = attn[k, h] (all n)
                b.y = attn_s[(kg + 1) * NH + th];        // B[2*sel+1,n]
                acc[i] = __builtin_amdgcn_wmma_f32_16x16x4_f32(
                    false, a, false, b, (short)0, acc[i], false, false);
            }
        }
        __syncthreads();
    }

    // D: every column identical; lane 0 holds M=0..7, lane 16 holds M=8..15.
    if ((lane & 15) == 0) {
        #pragma unroll
        for (int i = 0; i < 4; ++i) {
            const int task  = wave * 4 + i;
            const int th    = task & 15;
            const int fbase = (task >> 4) * 16 + sel * 8;
            #pragma unroll
            for (int rr = 0; rr < 8; ++rr) {
                out[(size_t)site * 512 + (fbase + rr) * NH + th] = acc[i][rr];
            }
        }
    }
}

extern "C" void kernel_launch(void* const* d_in, const int* in_sizes, int n_in,
                              void* d_out, int out_size, void* d_ws, size_t ws_size,
                              hipStream_t stream) {
    const float* beta = (const float*)d_in[0];
    const float* sa   = (const float*)d_in[1];
    const float* ac   = (const float*)d_in[2];
    const float* no   = (const float*)d_in[3];
    const float* gw   = (const float*)d_in[4];
    float* out = (float*)d_out;

    const int sites = in_sizes[0] / (NFG * NH);       // B*L = 4096
    float* out_attn = out + (size_t)sites * (NFO * NH);

    attn_layer_kernel<<<sites, 256, 0, stream>>>(beta, sa, ac, no, gw, out, out_attn);
}